// MMD_loss_14113262535030
// MI455X (gfx1250) — compile-verified
//
#include <hip/hip_runtime.h>

// MMD-RBF fused kernel for MI455X (gfx1250, wave32).
// Phase 1 (one workgroup per batch): stream U (4096x256 f32) through a
// double-buffered LDS stage in 32-row chunks, accumulate G = U^T U with
// V_WMMA_F32_16X16X4_F32 (16 waves, each owning a 4x4 block of 16x16 tiles ->
// 16 WMMAs per k-step), then fuse diagonal extraction, bandwidth computation
// and the 5-scale exp pass in-register/LDS. Per-batch signed kernel sum -> ws.
// Phase 2: fixed-order 64-element sum -> scalar loss (deterministic).

typedef __attribute__((ext_vector_type(2))) float v2f;
typedef __attribute__((ext_vector_type(4))) float f4;
typedef __attribute__((ext_vector_type(8))) float v8f;

#define B_   64
#define D_   4096
#define S_   128
#define N_   256        // 2S
#define KC   32         // k-chunk rows staged in LDS
#define LDST 272        // 256 + 16 floats pad: row+2 shifts banks by 32 -> no half-wave conflict
#define NCH  (D_ / KC)  // 128 chunks
#define BUFSZ (KC * LDST)

__global__ __launch_bounds__(512)
void mmd_batch_kernel(const float* __restrict__ src, const float* __restrict__ tgt,
                      float* __restrict__ partial)
{
    __shared__ float stage[2 * BUFSZ];            // 69,632 B (double buffered)
    const int tid     = threadIdx.x;
    const int lane    = tid & 31;
    const int wave    = tid >> 5;                 // 16 waves
    const int b       = blockIdx.x;
    const int bi      = wave >> 2;                // tile-block row (0..3)
    const int bj      = wave & 3;                 // tile-block col (0..3)
    const int halfsel = lane >> 4;                // 0: lanes 0-15, 1: lanes 16-31
    const int lcol    = lane & 15;

    // Column index into a staged U row for each owned tile (A and B fragments
    // of U^T U share the same lane layout per ISA 7.12.2).
    int rowCol[4], colCol[4];
#pragma unroll
    for (int a = 0; a < 4; ++a) {
        rowCol[a] = (bi * 4 + a) * 16 + lcol;
        colCol[a] = (bj * 4 + a) * 16 + lcol;
    }

    v8f acc[4][4];
#pragma unroll
    for (int a = 0; a < 4; ++a)
#pragma unroll
        for (int c = 0; c < 4; ++c)
            acc[a][c] = (v8f){0.f, 0.f, 0.f, 0.f, 0.f, 0.f, 0.f, 0.f};

    const size_t base = (size_t)b * D_ * S_;

    // Stage KC x 256 floats: row d -> [src[b,d,0:128] | tgt[b,d,0:128]]
    // Streaming data (each byte consumed once) -> non-temporal loads.
    auto load_chunk = [&](int chunk, float* buf) {
        const int d0 = chunk * KC;
#pragma unroll
        for (int i = 0; i < 4; ++i) {
            int idx = tid + i * 512;              // 2048 float4 transfers
            int row = idx >> 6;
            int q   = idx & 63;
            const float* p = (q < 32)
                ? (src + base + (size_t)(d0 + row) * S_ + q * 4)
                : (tgt + base + (size_t)(d0 + row) * S_ + (q - 32) * 4);
            f4 v = __builtin_nontemporal_load((const f4*)p);
            int c0 = (q < 32) ? q * 4 : 128 + (q - 32) * 4;
            *(f4*)&buf[row * LDST + c0] = v;
        }
    };

    load_chunk(0, stage);
    int pb = 0;
    for (int chunk = 0; chunk < NCH; ++chunk) {
        __syncthreads();   // staging of buffer pb complete for all waves
        if (chunk + 1 < NCH)
            load_chunk(chunk + 1, stage + (pb ^ 1) * BUFSZ);  // overlap with WMMA below
        const float* cur = stage + pb * BUFSZ;

#pragma unroll
        for (int kk = 0; kk < KC / 4; ++kk) {
            const int kb = kk * 4 + halfsel * 2;  // lanes 0-15: K=0,1 ; 16-31: K=2,3
            const float* p0 = &cur[kb * LDST];
            const float* p1 = &cur[(kb + 1) * LDST];
            v2f fr[4], fc[4];
#pragma unroll
            for (int a = 0; a < 4; ++a) {
                fr[a].x = p0[rowCol[a]];  fr[a].y = p1[rowCol[a]];
                fc[a].x = p0[colCol[a]];  fc[a].y = p1[colCol[a]];
            }
#pragma unroll
            for (int a = 0; a < 4; ++a)
#pragma unroll
                for (int c = 0; c < 4; ++c)
                    acc[a][c] = __builtin_amdgcn_wmma_f32_16x16x4_f32(
                        false, fr[a], false, fc[c], (short)0, acc[a][c], false, false);
        }
        pb ^= 1;
    }
    __syncthreads();

    // ---- fused epilogue: sq (diag), bandwidth, 5-scale exp, signed quadrant sum
    float* sqv = stage;          // [256]
    float* red = stage + 256;    // [512]
    float* ibw = stage + 1024;   // [5]

    // diagonal of G -> sq[]  (C/D layout: VGPR r, lanes<16 -> M=r, lanes>=16 -> M=r+8)
#pragma unroll
    for (int a = 0; a < 4; ++a)
#pragma unroll
        for (int c = 0; c < 4; ++c) {
            int ti = bi * 4 + a, tj = bj * 4 + c;
            if (ti == tj) {
#pragma unroll
                for (int r = 0; r < 8; ++r) {
                    int M = r + halfsel * 8;
                    if (M == lcol) sqv[ti * 16 + M] = acc[a][c][r];
                }
            }
        }

    // per-thread partial of sum(G)
    float sg = 0.f;
#pragma unroll
    for (int a = 0; a < 4; ++a)
#pragma unroll
        for (int c = 0; c < 4; ++c)
#pragma unroll
            for (int r = 0; r < 8; ++r) sg += acc[a][c][r];
    red[tid] = sg;
    __syncthreads();
    for (int s = 256; s > 0; s >>= 1) {
        if (tid < s) red[tid] += red[tid + s];
        __syncthreads();
    }

    if (tid == 0) {
        float ssq = 0.f;
        for (int n = 0; n < N_; ++n) ssq += sqv[n];
        // sum(L2) = 512*sum(diag) - 2*sum(G);  bw = sumL2/(n^2-n)/kernel_mul^2
        float bw = (512.0f * ssq - 2.0f * red[0]) * (1.0f / 65280.0f) * 0.25f;
        float sc = 1.0f;
        for (int k = 0; k < 5; ++k) { ibw[k] = 1.0f / (bw * sc); sc *= 2.0f; }
    }
    __syncthreads();

    float ib[5];
#pragma unroll
    for (int k = 0; k < 5; ++k) ib[k] = ibw[k];

    float loc = 0.f;
#pragma unroll
    for (int a = 0; a < 4; ++a)
#pragma unroll
        for (int c = 0; c < 4; ++c) {
            int Nn = (bj * 4 + c) * 16 + lcol;
            float sqN = sqv[Nn];
            bool nLow = Nn < S_;
#pragma unroll
            for (int r = 0; r < 8; ++r) {
                int Mm = (bi * 4 + a) * 16 + r + halfsel * 8;
                float l2 = sqv[Mm] + sqN - 2.0f * acc[a][c][r];
                float s = 0.f;
#pragma unroll
                for (int k = 0; k < 5; ++k) s += __expf(-l2 * ib[k]);
                loc += ((Mm < S_) == nLow) ? s : -s;
            }
        }

    __syncthreads();
    red[tid] = loc;
    __syncthreads();
    for (int s = 256; s > 0; s >>= 1) {
        if (tid < s) red[tid] += red[tid + s];
        __syncthreads();
    }
    if (tid == 0) partial[b] = red[0];
}

__global__ void mmd_finalize(const float* __restrict__ partial, float* __restrict__ out)
{
    if (threadIdx.x == 0 && blockIdx.x == 0) {
        float s = 0.f;
        for (int b = 0; b < B_; ++b) s += partial[b];   // fixed order -> deterministic
        out[0] = s * (1.0f / (64.0f * 16384.0f));       // /B and /(128*128) per quadrant
    }
}

extern "C" void kernel_launch(void* const* d_in, const int* in_sizes, int n_in,
                              void* d_out, int out_size, void* d_ws, size_t ws_size,
                              hipStream_t stream)
{
    const float* src = (const float*)d_in[0];
    const float* tgt = (const float*)d_in[1];
    float* out = (float*)d_out;
    float* ws  = (float*)d_ws;   // 64 floats of per-batch partials

    mmd_batch_kernel<<<dim3(B_), dim3(512), 0, stream>>>(src, tgt, ws);
    mmd_finalize<<<dim3(1), dim3(32), 0, stream>>>(ws, out);
}